// NeuralSelectionSort_33681133535757
// MI455X (gfx1250) — compile-verified
//
#include <hip/hip_runtime.h>
#include <hip/hip_bf16.h>

// ---------------------------------------------------------------------------
// Model constants (fixed by the reference)
// ---------------------------------------------------------------------------
#define BB  4
#define SS  2048
#define DD  64
#define HH  4
#define HD  16
#define FFD 256
#define LL  2

typedef float v2f __attribute__((ext_vector_type(2)));
typedef float v8f __attribute__((ext_vector_type(8)));

__device__ __forceinline__ v8f wmma_f32(v2f a, v2f b, v8f c) {
  // D = A(16x4, f32) * B(4x16, f32) + C(16x16, f32)
  return __builtin_amdgcn_wmma_f32_16x16x4_f32(false, a, false, b, (short)0, c,
                                               false, false);
}

// ---------------------------------------------------------------------------
// Embedding gather: h[b,s,:] = emb[x[b,s],:]
// ---------------------------------------------------------------------------
__global__ void k_embed(const int* __restrict__ x, const float* __restrict__ emb,
                        float* __restrict__ h) {
  int i = blockIdx.x * blockDim.x + threadIdx.x;
  if (i >= BB * SS * DD) return;
  int row = i >> 6;
  int d = i & 63;
  h[i] = emb[x[row] * DD + d];
}

// ---------------------------------------------------------------------------
// Generic WMMA GEMM:  C[m][n] = sum_k A[m][k] * W[n][k] + bias[n]
//   mode 0: plain store to C (ldc = N)
//   mode 1: GELU(exact) then store
//   mode 2: scatter to q/k/v  (N must be 192; q/k/v are (B,H,S,HD))
// One wave computes one 16x16 tile with K/4 fp32 WMMA ops.
// ---------------------------------------------------------------------------
__global__ void k_gemm(const float* __restrict__ A, const float* __restrict__ W,
                       const float* __restrict__ bias, float* __restrict__ C,
                       int M, int N, int K, int mode,
                       float* __restrict__ qp, float* __restrict__ kp,
                       float* __restrict__ vp) {
  int wid = threadIdx.x >> 5;
  int lane = threadIdx.x & 31;
  int tile = blockIdx.x * (blockDim.x >> 5) + wid;
  int tilesN = N >> 4;
  int tiles = (M >> 4) * tilesN;
  if (tile >= tiles) return;
  int tm = tile / tilesN;
  int tn = tile % tilesN;

  int l = lane & 15;
  bool hi = lane >= 16;
  int off = hi ? 2 : 0;

  const float* arow = A + (size_t)(tm * 16 + l) * K;
  const float* wrow = W + (size_t)(tn * 16 + l) * K;

  v8f acc = {};
  for (int k0 = 0; k0 < K; k0 += 4) {
    v2f a = *(const v2f*)(arow + k0 + off);
    v2f b = *(const v2f*)(wrow + k0 + off);
    acc = wmma_f32(a, b, acc);
  }

  int n = tn * 16 + l;
  float bv = bias[n];
#pragma unroll
  for (int r = 0; r < 8; ++r) {
    int m = tm * 16 + r + (hi ? 8 : 0);
    float val = acc[r] + bv;
    if (mode == 0) {
      C[(size_t)m * N + n] = val;
    } else if (mode == 1) {
      C[(size_t)m * N + n] = 0.5f * val * (1.0f + erff(val * 0.70710678118654752f));
    } else {
      // QKV scatter: n in [0,192): sel*64 + h*16 + d ; m = b*S + s
      int sel = n >> 6;
      int hh = (n >> 4) & 3;
      int d = n & 15;
      int b = m >> 11;
      int s = m & (SS - 1);
      float* dst = (sel == 0) ? qp : ((sel == 1) ? kp : vp);
      dst[(((size_t)(b * HH + hh)) * SS + s) * HD + d] = val;
    }
  }
}

// ---------------------------------------------------------------------------
// RoPE, in place on q and k, (B,H,S,HD) layout.  One thread per (tensor,b,h,s,j)
// pair j in [0,8): rotates (j, j+8).
// ---------------------------------------------------------------------------
__global__ void k_rope(float* __restrict__ q, float* __restrict__ k) {
  const int NPT = BB * HH * SS * 8;
  int i = blockIdx.x * blockDim.x + threadIdx.x;
  if (i >= 2 * NPT) return;
  float* base = (i < NPT) ? q : k;
  int t = (i < NPT) ? i : (i - NPT);
  int j = t & 7;
  int bhs = t >> 3;
  int s = bhs & (SS - 1);
  float inv = __powf(10000.0f, -(float)j * (1.0f / 8.0f));
  float ang = (float)s * inv;
  float cv = __cosf(ang), sv = __sinf(ang);
  float* p = base + (size_t)bhs * HD;
  float v1 = p[j];
  float v2 = p[j + 8];
  p[j] = v1 * cv - v2 * sv;
  p[j + 8] = v2 * cv + v1 * sv;
}

// ---------------------------------------------------------------------------
// Flash attention. One wave per (b, h, 16-query tile); streams keys in 16-col
// tiles with online softmax. Scores computed transposed (S^T = K * Q^T) so a
// lane holds a fixed query; partner lane (xor 16) holds the other 8 keys.
// Output written directly into (B,S,D) with d = h*16 + hd.
// ---------------------------------------------------------------------------
__global__ void k_flashattn(const float* __restrict__ q, const float* __restrict__ k,
                            const float* __restrict__ v, float* __restrict__ out) {
  int wid = threadIdx.x >> 5;
  int lane = threadIdx.x & 31;
  int unit = blockIdx.x * (blockDim.x >> 5) + wid;
  if (unit >= BB * HH * (SS / 16)) return;
  int b = unit >> 9;          // H * (S/16) = 512
  int rem = unit & 511;
  int h = rem >> 7;           // S/16 = 128
  int qt = rem & 127;

  int l = lane & 15;
  bool hi = lane >= 16;
  int off = hi ? 2 : 0;

  const float* qbase = q + ((size_t)(b * HH + h)) * SS * HD;
  const float* kbase = k + ((size_t)(b * HH + h)) * SS * HD;
  const float* vbase = v + ((size_t)(b * HH + h)) * SS * HD;

  // Preload Q^T B-fragments: lane (N=l, half) needs Q[N][4c+off], Q[N][4c+off+1]
  const float* qrow = qbase + (size_t)(qt * 16 + l) * HD;
  v2f qf[4];
#pragma unroll
  for (int c = 0; c < 4; ++c) qf[c] = *(const v2f*)(qrow + 4 * c + off);

  v8f o = {};
  float m = -1e30f;
  float lsum = 0.0f;

  for (int kb = 0; kb < SS; kb += 16) {
    // ---- scores tile S^T = K * Q^T (A = K rows, B = Q^T) ----
    const float* krow = kbase + (size_t)(kb + l) * HD;
    v8f sc = {};
#pragma unroll
    for (int c = 0; c < 4; ++c) {
      v2f a = *(const v2f*)(krow + 4 * c + off);
      sc = wmma_f32(a, qf[c], sc);
    }
    // lane holds scores for query q=l, keys kb + r + 8*hi  (r = 0..7)
    float p[8];
    float tmax = -1e30f;
#pragma unroll
    for (int r = 0; r < 8; ++r) {
      float s0 = sc[r] * 0.25f;   // HD^-0.5
      p[r] = s0;
      tmax = fmaxf(tmax, s0);
    }
    tmax = fmaxf(tmax, __shfl_xor(tmax, 16));
    float mnew = fmaxf(m, tmax);
    float f = __expf(m - mnew);
    float tsum = 0.0f;
#pragma unroll
    for (int r = 0; r < 8; ++r) {
      p[r] = __expf(p[r] - mnew);
      tsum += p[r];
    }
    tsum += __shfl_xor(tsum, 16);
    lsum = lsum * f + tsum;
    m = mnew;
#pragma unroll
    for (int r = 0; r < 8; ++r) o[r] *= f;

    // ---- O^T += V^T * P^T  (A = V^T fragment, B = P^T from p regs) ----
#pragma unroll
    for (int c = 0; c < 4; ++c) {
      int kk = kb + 4 * c + off;
      v2f a;
      a.x = vbase[(size_t)kk * HD + l];
      a.y = vbase[(size_t)(kk + 1) * HD + l];
      v2f bf;
      if (c < 2) {
        float t0 = __shfl_xor(p[4 * c + 2], 16);
        float t1 = __shfl_xor(p[4 * c + 3], 16);
        bf.x = hi ? t0 : p[4 * c];
        bf.y = hi ? t1 : p[4 * c + 1];
      } else {
        int j = 4 * c - 8;
        float t0 = __shfl_xor(p[j], 16);
        float t1 = __shfl_xor(p[j + 1], 16);
        bf.x = hi ? p[j + 2] : t0;
        bf.y = hi ? p[j + 3] : t1;
      }
      o = wmma_f32(a, bf, o);
    }
  }

  // O^T C-layout: lane (q=l) VGPR r holds O[q][r + 8*hi]  -> 8 consecutive d
  float inv_l = 1.0f / lsum;
  float* orow = out + ((size_t)(b * SS + qt * 16 + l)) * DD + h * HD + (hi ? 8 : 0);
#pragma unroll
  for (int r = 0; r < 8; ++r) orow[r] = o[r] * inv_l;
}

// ---------------------------------------------------------------------------
// Fused (optional residual) + LayerNorm over D=64. One wave per row.
// ---------------------------------------------------------------------------
__global__ void k_addln(const float* __restrict__ res, const float* __restrict__ y,
                        const float* __restrict__ sc, const float* __restrict__ bi,
                        float* __restrict__ out, int rows, int has_res) {
  int wid = (blockIdx.x * blockDim.x + threadIdx.x) >> 5;
  int lane = threadIdx.x & 31;
  if (wid >= rows) return;
  const float* yr = y + (size_t)wid * DD;
  float v0 = yr[lane];
  float v1 = yr[lane + 32];
  if (has_res) {
    const float* xr = res + (size_t)wid * DD;
    v0 += xr[lane];
    v1 += xr[lane + 32];
  }
  float s = v0 + v1;
  float sq = v0 * v0 + v1 * v1;
#pragma unroll
  for (int d = 1; d < 32; d <<= 1) {
    s += __shfl_xor(s, d);
    sq += __shfl_xor(sq, d);
  }
  float mean = s * (1.0f / 64.0f);
  float var = sq * (1.0f / 64.0f) - mean * mean;
  float rstd = rsqrtf(var + 1e-5f);
  float* orow = out + (size_t)wid * DD;
  orow[lane] = (v0 - mean) * rstd * sc[lane] + bi[lane];
  orow[lane + 32] = (v1 - mean) * rstd * sc[lane + 32] + bi[lane + 32];
}

// ---------------------------------------------------------------------------
// enc_mod = h + mask * sel_emb  (in place)
// ---------------------------------------------------------------------------
__global__ void k_selmod(float* __restrict__ h, const int* __restrict__ mask,
                         const float* __restrict__ sel) {
  int i = blockIdx.x * blockDim.x + threadIdx.x;
  if (i >= BB * SS * DD) return;
  int row = i >> 6;
  int d = i & 63;
  h[i] += (float)mask[row] * sel[d];
}

// ---------------------------------------------------------------------------
// context[b][d] = sum_s enc_mod * (1-mask) / max(count,1)
// ---------------------------------------------------------------------------
__global__ void k_context(const float* __restrict__ h, const int* __restrict__ mask,
                          float* __restrict__ ctx) {
  int b = blockIdx.x;
  int d = threadIdx.x;
  float acc = 0.0f, cnt = 0.0f;
  for (int s = 0; s < SS; ++s) {
    float rm = 1.0f - (float)mask[b * SS + s];
    acc += h[((size_t)(b * SS + s)) * DD + d] * rm;
    cnt += rm;
  }
  ctx[b * DD + d] = acc / fmaxf(cnt, 1.0f);
}

// ---------------------------------------------------------------------------
// query[b][n] = cq_w[n][:] . ctx[b][:] + cq_b[n]
// ---------------------------------------------------------------------------
__global__ void k_query(const float* __restrict__ ctx, const float* __restrict__ w,
                        const float* __restrict__ bi, float* __restrict__ qry) {
  int b = blockIdx.x;
  int n = threadIdx.x;
  float acc = bi[n];
  for (int kk = 0; kk < DD; ++kk) acc += w[n * DD + kk] * ctx[b * DD + kk];
  qry[b * DD + n] = acc;
}

// ---------------------------------------------------------------------------
// logits[b][s] = masked scaled dot(query[b], keys[b,s])
// ---------------------------------------------------------------------------
__global__ void k_logits(const float* __restrict__ qry, const float* __restrict__ keys,
                         const int* __restrict__ mask, float* __restrict__ out) {
  int i = blockIdx.x * blockDim.x + threadIdx.x;
  if (i >= BB * SS) return;
  if (mask[i]) {
    out[i] = -__builtin_inff();
    return;
  }
  int b = i / SS;
  float acc = 0.0f;
  for (int d = 0; d < DD; ++d) acc += qry[b * DD + d] * keys[(size_t)i * DD + d];
  out[i] = acc * 2.8284271247461903f;  // sqrt(S/4)/sqrt(D)
}

// ---------------------------------------------------------------------------
// Host orchestration
// ---------------------------------------------------------------------------
extern "C" void kernel_launch(void* const* d_in, const int* in_sizes, int n_in,
                              void* d_out, int out_size, void* d_ws, size_t ws_size,
                              hipStream_t stream) {
  (void)in_sizes; (void)n_in; (void)out_size; (void)ws_size;

  const int* x_in = (const int*)d_in[0];
  const int* mask = (const int*)d_in[1];
  const float* emb = (const float*)d_in[2];
  const float* enc_ln_s = (const float*)d_in[27];
  const float* enc_ln_b = (const float*)d_in[28];
  const float* cq_w = (const float*)d_in[29];
  const float* cq_b = (const float*)d_in[30];
  const float* pk_w = (const float*)d_in[31];
  const float* pk_b = (const float*)d_in[32];
  const float* sel_emb = (const float*)d_in[33];

  float* ws = (float*)d_ws;
  const size_t T = (size_t)BB * SS * DD;  // 512K floats
  float* h    = ws;                // [0, T)
  float* q    = ws + T;            // [T, 2T)     (aliased by ff after attn)
  float* k    = ws + 2 * T;
  float* v    = ws + 3 * T;
  float* attn = ws + 4 * T;
  float* ff   = ws + T;            // B*S*FF = 4T floats, overlaps q/k/v/attn
  float* y    = ws + 5 * T;
  float* keys = ws + T;            // reuse q region post-layers
  float* ctx  = ws + 6 * T;
  float* qry  = ws + 6 * T + 256;

  const int THREADS = 256;
  const int ROWS = BB * SS;  // 8192

  // token embedding
  k_embed<<<(BB * SS * DD + THREADS - 1) / THREADS, THREADS, 0, stream>>>(x_in, emb, h);

  for (int layer = 0; layer < LL; ++layer) {
    int lb = 3 + 12 * layer;
    const float* qkv_w = (const float*)d_in[lb + 0];
    const float* qkv_b = (const float*)d_in[lb + 1];
    const float* out_w = (const float*)d_in[lb + 2];
    const float* out_b = (const float*)d_in[lb + 3];
    const float* ln1_s = (const float*)d_in[lb + 4];
    const float* ln1_b = (const float*)d_in[lb + 5];
    const float* ln2_s = (const float*)d_in[lb + 6];
    const float* ln2_b = (const float*)d_in[lb + 7];
    const float* ff1_w = (const float*)d_in[lb + 8];
    const float* ff1_b = (const float*)d_in[lb + 9];
    const float* ff2_w = (const float*)d_in[lb + 10];
    const float* ff2_b = (const float*)d_in[lb + 11];

    // QKV projection (M=8192, N=192, K=64) with scatter into (B,H,S,HD)
    {
      int tiles = (ROWS / 16) * (192 / 16);
      k_gemm<<<(tiles + 7) / 8, THREADS, 0, stream>>>(h, qkv_w, qkv_b, nullptr,
                                                      ROWS, 192, DD, 2, q, k, v);
    }
    // RoPE on q and k
    k_rope<<<(2 * BB * HH * SS * 8 + THREADS - 1) / THREADS, THREADS, 0, stream>>>(q, k);
    // flash attention -> attn (B,S,D)
    k_flashattn<<<(BB * HH * (SS / 16) + 7) / 8, THREADS, 0, stream>>>(q, k, v, attn);
    // out projection (M=8192, N=64, K=64) -> y
    {
      int tiles = (ROWS / 16) * (DD / 16);
      k_gemm<<<(tiles + 7) / 8, THREADS, 0, stream>>>(attn, out_w, out_b, y,
                                                      ROWS, DD, DD, 0,
                                                      nullptr, nullptr, nullptr);
    }
    // h = LN(h + y)
    k_addln<<<(ROWS + 7) / 8, THREADS, 0, stream>>>(h, y, ln1_s, ln1_b, h, ROWS, 1);
    // FF1 + GELU (M=8192, N=256, K=64) -> ff
    {
      int tiles = (ROWS / 16) * (FFD / 16);
      k_gemm<<<(tiles + 7) / 8, THREADS, 0, stream>>>(h, ff1_w, ff1_b, ff,
                                                      ROWS, FFD, DD, 1,
                                                      nullptr, nullptr, nullptr);
    }
    // FF2 (M=8192, N=64, K=256) -> y
    {
      int tiles = (ROWS / 16) * (DD / 16);
      k_gemm<<<(tiles + 7) / 8, THREADS, 0, stream>>>(ff, ff2_w, ff2_b, y,
                                                      ROWS, DD, FFD, 0,
                                                      nullptr, nullptr, nullptr);
    }
    // h = LN(h + y)
    k_addln<<<(ROWS + 7) / 8, THREADS, 0, stream>>>(h, y, ln2_s, ln2_b, h, ROWS, 1);
  }

  // final encoder LN (no residual), in place
  k_addln<<<(ROWS + 7) / 8, THREADS, 0, stream>>>(nullptr, h, enc_ln_s, enc_ln_b,
                                                  h, ROWS, 0);
  // enc_mod = h + mask * sel_emb (in place)
  k_selmod<<<(BB * SS * DD + THREADS - 1) / THREADS, THREADS, 0, stream>>>(h, mask, sel_emb);
  // masked mean context per batch
  k_context<<<BB, DD, 0, stream>>>(h, mask, ctx);
  // query = cq(context)
  k_query<<<BB, DD, 0, stream>>>(ctx, cq_w, cq_b, qry);
  // keys = pk(enc_mod)  (M=8192, N=64, K=64)
  {
    int tiles = (ROWS / 16) * (DD / 16);
    k_gemm<<<(tiles + 7) / 8, THREADS, 0, stream>>>(h, pk_w, pk_b, keys,
                                                    ROWS, DD, DD, 0,
                                                    nullptr, nullptr, nullptr);
  }
  // logits
  k_logits<<<(BB * SS + THREADS - 1) / THREADS, THREADS, 0, stream>>>(
      qry, keys, mask, (float*)d_out);
}